// AngularTermsANI1_15625091023137
// MI455X (gfx1250) — compile-verified
//
#include <hip/hip_runtime.h>
#include <math.h>

// ANI-1x angular terms, MI455X (gfx1250).
// Memory-bound streaming kernel: 304 MB traffic -> ~13 us floor @ 23.3 TB/s.
// Strategy: 1 thread per pair, NT b96 loads, all-register transcendental math,
// NT b128 streaming stores (output 256 MB > 192 MB L2, so bypass-friendly).

typedef float v4f __attribute__((ext_vector_type(4)));

#define ANI_CUTOFF 3.5f

// Param block layout in d_ws (floats):
//  [0..7]   cos(ShfZ[s])
//  [8..15]  sin(ShfZ[s])
//  [16..19] ShfA[a]
//  [20]     EtaA
//  [21]     Zeta
__global__ void ani_angular_setup(const float* __restrict__ EtaA,
                                  const float* __restrict__ Zeta,
                                  const float* __restrict__ ShfA,
                                  const float* __restrict__ ShfZ,
                                  float* __restrict__ params) {
  int t = threadIdx.x;  // single wave32
  if (t < 8) {
    float s, c;
    __sincosf(ShfZ[t], &s, &c);
    params[t]     = c;
    params[8 + t] = s;
  }
  if (t < 4) params[16 + t] = ShfA[t];
  if (t == 0) { params[20] = EtaA[0]; params[21] = Zeta[0]; }
}

__device__ __forceinline__ float pow32(float x) {
  float x2  = x  * x;
  float x4  = x2 * x2;
  float x8  = x4 * x4;
  float x16 = x8 * x8;
  return x16 * x16;
}

__global__ __launch_bounds__(256)
void ani_angular_main(const float* __restrict__ vec,     // (2, P, 3)
                      const float* __restrict__ params,  // d_ws param block
                      float* __restrict__ out,           // (P, 32)
                      int P) {
  int p = blockIdx.x * blockDim.x + threadIdx.x;
  if (p >= P) return;

  const float* v0p = vec + (size_t)3 * (size_t)p;
  const float* v1p = vec + (size_t)3 * (size_t)P + (size_t)3 * (size_t)p;

  // Streamed once: non-temporal; adjacent scalars merge into global_load_b96 nt.
  float x0 = __builtin_nontemporal_load(v0p + 0);
  float y0 = __builtin_nontemporal_load(v0p + 1);
  float z0 = __builtin_nontemporal_load(v0p + 2);
  float x1 = __builtin_nontemporal_load(v1p + 0);
  float y1 = __builtin_nontemporal_load(v1p + 1);
  float z1 = __builtin_nontemporal_load(v1p + 2);

  float d0 = sqrtf(x0 * x0 + y0 * y0 + z0 * z0);
  float d1 = sqrtf(x1 * x1 + y1 * y1 + z1 * z1);
  float dot = x0 * x1 + y0 * y1 + z0 * z1;

  // t = 0.95 * cos(angle); |t| <= 0.95 so sin(arccos t) = sqrt(1-t^2) is safe.
  float denom = fmaxf(d0 * d1, 1e-10f);
  float t = 0.95f * __fdividef(dot, denom);
  float sint = sqrtf(fmaxf(1.0f - t * t, 0.0f));

  // Cosine cutoff product and pre-factor 2*fc0*fc1.
  const float k = 3.14159265358979323846f / ANI_CUTOFF;
  float fc0 = 0.5f * __cosf(d0 * k) + 0.5f;
  float fc1 = 0.5f * __cosf(d1 * k) + 0.5f;
  float pre = 2.0f * fc0 * fc1;
  float dmean = 0.5f * (d0 + d1);

  // Uniform addresses -> scalar (SMEM) loads, L2-resident after setup kernel.
  float eta  = params[20];
  float zeta = params[21];

  // Radial factor folded with the 2*fcj prefactor: 4 values.
  float f2[4];
#pragma unroll
  for (int a = 0; a < 4; ++a) {
    float dd = dmean - params[16 + a];
    f2[a] = pre * __expf(-eta * dd * dd);
  }

  // Angular factor: cos(theta - ShfZ) = t*cosZ + sin(theta)*sinZ (no arccos).
  // zeta == 32 (canonical): exact 5-squaring power. Generic fallback uses
  // exp(zeta*log(base)): base in [0,1], log(0) = -inf -> exp -> 0, correct
  // limit, and only 2 trans ops instead of the inlined OCML pow polynomial.
  float f1[8];
  if (zeta == 32.0f) {
#pragma unroll
    for (int s = 0; s < 8; ++s) {
      float base = 0.5f + 0.5f * (t * params[s] + sint * params[8 + s]);
      f1[s] = pow32(base);
    }
  } else {
#pragma unroll
    for (int s = 0; s < 8; ++s) {
      float base = 0.5f + 0.5f * (t * params[s] + sint * params[8 + s]);
      f1[s] = __expf(zeta * __logf(base));
    }
  }

  // 32 outputs = outer product f2[a] * f1[s]; 128 contiguous bytes per thread,
  // written as 8 non-temporal global_store_b128 (streaming, no L2 pollution).
  float* o = out + (size_t)p * 32;
#pragma unroll
  for (int a = 0; a < 4; ++a) {
#pragma unroll
    for (int s = 0; s < 8; s += 4) {
      v4f v;
      v.x = f2[a] * f1[s + 0];
      v.y = f2[a] * f1[s + 1];
      v.z = f2[a] * f1[s + 2];
      v.w = f2[a] * f1[s + 3];
      __builtin_nontemporal_store(v, (v4f*)(o + a * 8 + s));
    }
  }
}

extern "C" void kernel_launch(void* const* d_in, const int* in_sizes, int n_in,
                              void* d_out, int out_size, void* d_ws, size_t ws_size,
                              hipStream_t stream) {
  const float* vec  = (const float*)d_in[0];  // (2, P, 3) f32
  const float* EtaA = (const float*)d_in[1];  // (1,)
  const float* Zeta = (const float*)d_in[2];  // (1,)
  const float* ShfA = (const float*)d_in[3];  // (4,)
  const float* ShfZ = (const float*)d_in[4];  // (8,)
  float* out = (float*)d_out;                 // (P, 32) f32
  float* params = (float*)d_ws;               // 22 floats of scratch

  int P = in_sizes[0] / 6;  // 2 * P * 3 flat elements

  ani_angular_setup<<<1, 32, 0, stream>>>(EtaA, Zeta, ShfA, ShfZ, params);

  const int block = 256;  // 8 wave32s per block
  const int grid = (P + block - 1) / block;
  ani_angular_main<<<grid, block, 0, stream>>>(vec, params, out, P);
}